// Predictor_82111184765047
// MI455X (gfx1250) — compile-verified
//
#include <hip/hip_runtime.h>
#include <hip/hip_bf16.h>

// CDNA5 / gfx1250: batched tiny-MLP chain using V_WMMA_F32_16X16X4_F32.
// M = 16 pixels per wave tile, N = layer outputs (zero-padded to 16),
// K = layer inputs (chunked by 4, zero-padded).
//
// Per-pixel ctx state is stored INTERLEAVED so each MLP's input is a
// contiguous prefix:  slot0=x0, slot(2m-1)=x_m, slot(2m)=e_{m-1}.
// The prep kernel permutes first-layer weight columns to match, so the
// main kernel's A-operand loads are branch-free aligned ds_load_b64.
// Hidden activations live in a neuron-major [16 neurons][16 pixels]
// buffer: WMMA D writeback = two ds_store_b128 per lane.
// Each wave grid-strides over ~8 tiles so the per-block weight staging
// (34KB L2->LDS) is amortized 8x.

typedef __attribute__((ext_vector_type(2))) float v2f;
typedef __attribute__((ext_vector_type(4))) float v4f;
typedef __attribute__((ext_vector_type(8))) float v8f;

#define N_LAYERS 48
#define WS_BIAS  768          // 48 layers * 16 padded bias floats
#define WS_TOTAL 8704         // + 124 K-chunks * 64 floats (B operand layout)

#define CTX_STRIDE 34         // dwords per pixel ctx row (even: b64-aligned; conflict-free)
#define HID_OFF    544        // 16*34 ctx rows, then 16x16 hidden buffer
#define WAVE_LDS   800        // dwords of sAct per wave
#define WAVES_PER_BLOCK 8
#define TILE_PX 16
#define TILES_PER_WAVE 8      // grid-stride factor (amortizes weight staging)

// Per-layer metadata (compile-time: layer loop is fully unrolled).
constexpr int LKIN[N_LAYERS] = {
    3,2, 5,3, 7,4,2, 9,5,2, 11,6,3, 13,7,3,
    15,8,4,2, 17,9,4,2, 19,10,5,2, 21,11,5,2,
    23,12,6,3, 25,13,6,3, 27,14,7,3, 29,15,7,3};
constexpr int LKOUT[N_LAYERS] = {
    2,1, 3,1, 4,2,1, 5,2,1, 6,3,1, 7,3,1,
    8,4,2,1, 9,4,2,1, 10,5,2,1, 11,5,2,1,
    12,6,3,1, 13,6,3,1, 14,7,3,1, 15,7,3,1};
constexpr int LKP[N_LAYERS] = {
    2,2, 3,3, 4,4,4, 5,5,5, 6,6,6, 7,7,7,
    8,8,8,8, 9,9,9,9, 10,10,10,10, 11,11,11,11,
    12,12,12,12, 13,13,13,13, 14,14,14,14, 15,15,15,15};

struct PtrPack { const float* p[96]; };   // 48 layers x (W, b)

static __device__ __forceinline__ void lds_wave_sync() {
    __builtin_amdgcn_wave_barrier();
    asm volatile("s_wait_dscnt 0" ::: "memory");
    __builtin_amdgcn_wave_barrier();
}

static __device__ __forceinline__ float leaky(float v) {
    return fmaxf(v, 0.01f * v);           // == leaky_relu(slope 0.01)
}

// ---------------------------------------------------------------------------
// Prep: marshal raw (W,b) into WMMA B-operand layout.
// Chunk entry e in [0,64): lane=e>>1, half=e&1, N=lane&15,
//   K = 4*chunk + (lane>>4)*2 + half   (low lanes K{4c,4c+1}, high {4c+2,4c+3})
// For first layers, column K maps through the interleaved-ctx permutation:
//   K==0 -> x0 (j=0); K odd -> x_{(K+1)/2}; K even -> e_{K/2-1} (j=kp+K/2-1).
// Zero-padded for N>=kout or K>=kin.
// ---------------------------------------------------------------------------
__global__ __launch_bounds__(256) void prep_weights(PtrPack pk, float* __restrict__ ws) {
    const int tid = threadIdx.x;
    int coff = WS_BIAS;
#pragma unroll
    for (int l = 0; l < N_LAYERS; ++l) {
        const int kin  = LKIN[l];
        const int kout = LKOUT[l];
        const int kp   = LKP[l];
        const int nch  = (kin + 3) >> 2;
        const bool first = (kin == 2 * kp - 1);
        const float* Wm = pk.p[2 * l];
        const float* bv = pk.p[2 * l + 1];
        if (tid < 16) ws[l * 16 + tid] = (tid < kout) ? bv[tid] : 0.0f;
        const int tot = nch * 64;
        for (int e = tid; e < tot; e += 256) {
            const int ch   = e >> 6;
            const int r    = e & 63;
            const int lane = r >> 1;
            const int half = r & 1;
            const int N = lane & 15;
            const int K = ch * 4 + ((lane >> 4) << 1) + half;
            int j = K;
            if (first) j = (K == 0) ? 0 : ((K & 1) ? ((K + 1) >> 1) : (kp + (K >> 1) - 1));
            ws[coff + e] = (N < kout && K < kin) ? Wm[N * kin + j] : 0.0f;
        }
        coff += tot;
    }
}

// ---------------------------------------------------------------------------
// Main kernel: one wave = 16 pixels per tile, several tiles per wave.
// ctx row per pixel (stride 34): [0]=x0, [2m-1]=x_m, [2m]=e_{m-1}, [31]=scratch.
// hidden buffer per wave: [neuron][pixel] 16x16 (rows >= kout are exact 0).
// ---------------------------------------------------------------------------
__global__ __launch_bounds__(256) void predictor_wmma(const float* __restrict__ x,
                                                      const float* __restrict__ ws,
                                                      float* __restrict__ out,
                                                      int ntiles) {
    __shared__ __align__(16) float sAct[WAVES_PER_BLOCK * WAVE_LDS];
    __shared__ __align__(16) float sW[WS_TOTAL];

    const int tid = threadIdx.x;
    for (int i = tid; i < WS_TOTAL / 4; i += 256)          // b128 copy, once per block
        ((v4f*)sW)[i] = ((const v4f*)ws)[i];
    __syncthreads();

    const int wave = tid >> 5;
    const int lane = tid & 31;
    const int hi   = lane >> 4;      // which K half this lane supplies
    const int nn   = lane & 15;      // N column (and A-operand pixel index)

    float* ctxB = &sAct[wave * WAVE_LDS];
    float* hidB = ctxB + HID_OFF;
    const float* myctx = ctxB + nn * CTX_STRIDE;

    const int wave_global = blockIdx.x * WAVES_PER_BLOCK + wave;
    const int nwaves      = gridDim.x * WAVES_PER_BLOCK;

    for (int tile = wave_global; tile < ntiles; tile += nwaves) {   // uniform per wave
        const long pixbase = (long)tile * TILE_PX;

        // Stage 16 pixels x 16 channels into interleaved slots, coalesced.
#pragma unroll
        for (int r = 0; r < 8; ++r) {
            const int g  = r * 32 + lane;
            const int p  = g >> 4, ch = g & 15;
            const int sl = ch ? (2 * ch - 1) : 0;
            ctxB[p * CTX_STRIDE + sl] = x[pixbase * 16 + g];
        }
        // Zero error slots (2,4,...,30) + scratch(31) so padded reads are finite.
#pragma unroll
        for (int r = 0; r < 8; ++r) {
            const int g  = r * 32 + lane;
            const int p  = g >> 4, i = g & 15;
            const int sl = (i == 15) ? 31 : (2 + 2 * i);
            ctxB[p * CTX_STRIDE + sl] = 0.0f;
        }
        lds_wave_sync();

        // e0 = x1 - x0
        if (lane < 16) {
            float* row = ctxB + lane * CTX_STRIDE;
            row[2] = row[1] - row[0];
        }
        lds_wave_sync();

        int coff = 0;
#pragma unroll
        for (int l = 0; l < N_LAYERS; ++l) {
            const int kin   = LKIN[l];
            const int kout  = LKOUT[l];
            const int kp    = LKP[l];
            const int nch   = (kin + 3) >> 2;
            const bool first = (kin == 2 * kp - 1);
            const bool last  = (kout == 1);

            if (!last) {
                const float bias = sW[l * 16 + nn];
                v8f c = {bias, bias, bias, bias, bias, bias, bias, bias};
                if (first) {
#pragma unroll
                    for (int ch = 0; ch < nch; ++ch) {
                        const int k0 = ch * 4 + hi * 2;
                        const v2f a = *(const v2f*)&myctx[k0];                 // ds_load_b64
                        const v2f b = *(const v2f*)&sW[WS_BIAS + coff + ch * 64 + lane * 2];
                        c = __builtin_amdgcn_wmma_f32_16x16x4_f32(
                                false, a, false, b, (short)0, c, false, false);
                    }
                } else {
#pragma unroll
                    for (int ch = 0; ch < nch; ++ch) {
                        const int k0 = ch * 4 + hi * 2;
                        v2f a;
                        a.x = hidB[k0 * 16 + nn];
                        a.y = hidB[(k0 + 1) * 16 + nn];
                        const v2f b = *(const v2f*)&sW[WS_BIAS + coff + ch * 64 + lane * 2];
                        c = __builtin_amdgcn_wmma_f32_16x16x4_f32(
                                false, a, false, b, (short)0, c, false, false);
                    }
                }
                // leaky ReLU, neuron-major writeback: lane -> hid[nn][hi*8 .. hi*8+7]
                v4f lo  = {leaky(c[0]), leaky(c[1]), leaky(c[2]), leaky(c[3])};
                v4f hi4 = {leaky(c[4]), leaky(c[5]), leaky(c[6]), leaky(c[7])};
                *(v4f*)&hidB[nn * 16 + hi * 8]     = lo;                        // ds_store_b128
                *(v4f*)&hidB[nn * 16 + hi * 8 + 4] = hi4;
                lds_wave_sync();
            } else {
                // kout==1, kin<=3: scalar dot beats a 16x16x4 WMMA. Weights come
                // from the zero-padded B chunk: (N=0,K=0/1/2) at entries 0,1,32.
                if (lane < 16) {
                    const float w0 = sW[WS_BIAS + coff + 0];
                    const float w1 = sW[WS_BIAS + coff + 1];
                    const float w2 = sW[WS_BIAS + coff + 32];
                    float pred = sW[l * 16 + 0]
                               + w0 * hidB[lane]
                               + w1 * hidB[16 + lane]
                               + w2 * hidB[32 + lane];
                    pred = fminf(1.0f, fmaxf(-1.0f, pred));
                    // delta = fmod(x_kp - pred + 1, 2) - 1 ; argument in [0,3]
                    float t = ctxB[lane * CTX_STRIDE + (2 * kp - 1)] - pred + 1.0f;
                    t = (t >= 2.0f) ? (t - 2.0f) : t;
                    ctxB[lane * CTX_STRIDE + 2 * kp] = t - 1.0f;   // e_{kp-1}
                }
                lds_wave_sync();
            }
            coff += nch * 64;
        }

        // Write 16 pixels x 15 errors (e_j at slot 2j+2), coalesced.
#pragma unroll
        for (int t = 0; t < 8; ++t) {
            const int idx = t * 32 + lane;
            if (idx < 240) {
                out[pixbase * 15 + idx] = ctxB[(idx / 15) * CTX_STRIDE + 2 + 2 * (idx % 15)];
            }
        }
        lds_wave_sync();   // tile's LDS reads done before next tile overwrites
    }
}

extern "C" void kernel_launch(void* const* d_in, const int* in_sizes, int n_in,
                              void* d_out, int out_size, void* d_ws, size_t ws_size,
                              hipStream_t stream) {
    (void)out_size; (void)ws_size;

    PtrPack pk;
    for (int i = 0; i < 96; ++i) {
        const int src = (i + 1 < n_in) ? (i + 1) : (n_in - 1);
        pk.p[i] = (const float*)d_in[src];
    }
    prep_weights<<<1, 256, 0, stream>>>(pk, (float*)d_ws);

    const int npix   = in_sizes[0] / 16;                  // 128*128*128
    const int ntiles = npix / TILE_PX;                    // 16 pixels per wave-tile
    int blocks = (ntiles + WAVES_PER_BLOCK * TILES_PER_WAVE - 1)
               / (WAVES_PER_BLOCK * TILES_PER_WAVE);      // ~8 tiles per wave
    if (blocks < 1) blocks = 1;
    predictor_wmma<<<blocks, 256, 0, stream>>>(
        (const float*)d_in[0], (const float*)d_ws, (float*)d_out, ntiles);
}